// ReplacementClassificationHead_71476845740454
// MI455X (gfx1250) — compile-verified
//
#include <hip/hip_runtime.h>

typedef __attribute__((ext_vector_type(2))) float v2f;
typedef __attribute__((ext_vector_type(8))) float v8f;

#define BS        32
#define SEQ       4096
#define DIM       1024
#define NPAIR     512
#define OUT_N     2
#define TILES_PER_BATCH (NPAIR / 16)          // 32
#define NTILES    (BS * TILES_PER_BATCH)      // 1024 tiles of 16 pairs
#define OUT_ELEMS (BS * NPAIR * OUT_N)        // 32768 output floats

// One K-chunk of the 16x(K)x16 f32 GEMM: per k-step, each lane feeds one
// contiguous float2 of A (gathered hidden row) and of B (W row) into
// v_wmma_f32_16x16x4_f32. A layout: lane m+16h holds A[m][k+2h..k+2h+1];
// B layout: lane n+16h holds B[k+2h..k+2h+1][n] = W[n][off+k+2h..+1].
// unroll 16 -> ~32 outstanding global_load_b64 per wave (LOADcnt max 63),
// enough bytes-in-flight to cover HBM latency at 8192 waves.
template <int KLEN>
__device__ __forceinline__ v8f dot_part(const float* __restrict__ ap,
                                        const float* __restrict__ bp,
                                        v8f acc) {
#pragma unroll 16
  for (int k = 0; k < KLEN; k += 4) {
    v2f av = *(const v2f*)(ap + k);
    v2f bv = *(const v2f*)(bp + k);
    acc = __builtin_amdgcn_wmma_f32_16x16x4_f32(
        /*neg_a=*/false, av, /*neg_b=*/false, bv,
        /*c_mod=*/(short)0, acc, /*reuse_a=*/false, /*reuse_b=*/false);
  }
  return acc;
}

// grid.x = NSEG * NTILES, block = 32 (one wave = one 16-pair tile, one K-seg).
template <int KLEN>
__global__ __launch_bounds__(32) void head_partial_kernel(
    const float* __restrict__ hidden,   // (32, 4096, 1024) f32
    const float* __restrict__ pooled,   // (32, 1024) f32
    const int*   __restrict__ pairs,    // (32, 512, 2) int32
    const float* __restrict__ W,        // (2, 3072) f32
    float* __restrict__ partial)        // (NSEG, 32768) f32
{
  const int lane = threadIdx.x;     // 0..31
  const int m    = lane & 15;       // A row / D column index
  const int half = lane >> 4;       // K sub-phase (0 -> k+0..1, 1 -> k+2..3)

  const int seg  = blockIdx.x / NTILES;
  const int tile = blockIdx.x % NTILES;
  const int b    = tile >> 5;             // batch
  const int p0   = (tile & 31) * 16;      // first pair of tile

  // Gather indices for this lane's matrix row.
  const int pairBase = ((b * NPAIR) + p0 + m) * OUT_N;
  const long long li = (long long)pairs[pairBase + 0];
  const long long ri = (long long)pairs[pairBase + 1];

  const int kbase = seg * KLEN + 2 * half;
  const float* aPooled = pooled + (size_t)b * DIM + kbase;
  const float* aLeft   = hidden + ((size_t)b * SEQ + li) * DIM + kbase;
  const float* aRight  = hidden + ((size_t)b * SEQ + ri) * DIM + kbase;

  // B: lane supplies W column n = m (lanes n>=2 duplicate column 0; only D
  // columns 0..1 are ever read, so the extra columns are don't-care).
  const int n = (m < OUT_N) ? m : 0;
  const float* wRow = W + (size_t)n * (3 * DIM) + kbase;

  v8f acc = {0.f, 0.f, 0.f, 0.f, 0.f, 0.f, 0.f, 0.f};
  acc = dot_part<KLEN>(aPooled, wRow,           acc);  // pooled . W[:,    0:1024]
  acc = dot_part<KLEN>(aLeft,   wRow + DIM,     acc);  // left   . W[:, 1024:2048]
  acc = dot_part<KLEN>(aRight,  wRow + 2 * DIM, acc);  // right  . W[:, 2048:3072]

  // D layout: VGPR r, lanes 0-15 -> row r col lane; lanes 16-31 -> row r+8.
  // Lanes with column index < 2 own the real outputs.
  if (m < OUT_N) {
    float* dst = partial + (size_t)seg * OUT_ELEMS +
                 ((size_t)(b * NPAIR + p0)) * OUT_N + m;
#pragma unroll
    for (int r = 0; r < 8; ++r)
      dst[(size_t)(half * 8 + r) * OUT_N] = acc[r];
  }
}

__global__ void head_reduce_kernel(const float* __restrict__ partial, int nseg,
                                   const float* __restrict__ bias,
                                   const float* __restrict__ ebias,
                                   float* __restrict__ out) {
  int i = blockIdx.x * blockDim.x + threadIdx.x;
  if (i >= OUT_ELEMS) return;
  float s = 0.f;
  for (int sgi = 0; sgi < nseg; ++sgi)
    s += partial[(size_t)sgi * OUT_ELEMS + i];
  out[i] = s + bias[i & 1] + ebias[i & 1];
}

extern "C" void kernel_launch(void* const* d_in, const int* in_sizes, int n_in,
                              void* d_out, int out_size, void* d_ws, size_t ws_size,
                              hipStream_t stream) {
  const float* hidden = (const float*)d_in[0];
  const float* pooled = (const float*)d_in[1];
  const int*   pairs  = (const int*)d_in[2];   // indices < 4096 fit in i32
  const float* W      = (const float*)d_in[3];
  const float* bias   = (const float*)d_in[4];
  const float* ebias  = (const float*)d_in[5];
  float* out = (float*)d_out;
  float* ws  = (float*)d_ws;

  const size_t seg_bytes = (size_t)OUT_ELEMS * sizeof(float);  // 128 KB
  if (ws_size >= 8 * seg_bytes) {
    // 8-way K split -> 8192 waves; each covers HBM latency with ~32
    // outstanding b64 loads (unroll 16).
    head_partial_kernel<DIM / 8>
        <<<dim3(8 * NTILES), dim3(32), 0, stream>>>(hidden, pooled, pairs, W, ws);
    head_reduce_kernel<<<(OUT_ELEMS + 255) / 256, 256, 0, stream>>>(
        ws, 8, bias, ebias, out);
  } else if (ws_size >= 4 * seg_bytes) {
    head_partial_kernel<DIM / 4>
        <<<dim3(4 * NTILES), dim3(32), 0, stream>>>(hidden, pooled, pairs, W, ws);
    head_reduce_kernel<<<(OUT_ELEMS + 255) / 256, 256, 0, stream>>>(
        ws, 4, bias, ebias, out);
  } else {
    head_partial_kernel<DIM>
        <<<dim3(NTILES), dim3(32), 0, stream>>>(hidden, pooled, pairs, W, ws);
    head_reduce_kernel<<<(OUT_ELEMS + 255) / 256, 256, 0, stream>>>(
        ws, 1, bias, ebias, out);
  }
}